// Attention_SWC_36575941492781
// MI455X (gfx1250) — compile-verified
//
#include <hip/hip_runtime.h>
#include <math.h>

// Problem constants (from reference)
#define N      32768            // samples per row per channel
#define WIN    100              // WINDOW*3-2
#define OUTC   32669            // N - WIN + 1
#define ROWS   512
#define TILE   4096             // outputs per block
#define HALO   (WIN - 1)        // 99
#define NLOAD  (TILE + HALO)    // 4195 inputs needed per tile
#define STRIDE 32672            // padded corr row stride (multiple of 32)

// LDS skew: per-thread windows start at multiples of 16; skewing by i>>4 turns
// the stride-16 lane pattern into stride-17 (coprime with 64 banks) -> conflict free.
#define SK(i) ((i) + ((i) >> 4))
#define SKMAX 4460              // SK(4194)=4456 max index accessed

#if defined(__gfx1250__) && __has_builtin(__builtin_amdgcn_global_load_async_to_lds_b32)
#define USE_ASYNC_LDS 1
// Builtin signature (per hipcc diagnostic): non-const int* in AS(1)/AS(3).
typedef __attribute__((address_space(1))) int g_int;
typedef __attribute__((address_space(3))) int l_int;
#endif

typedef float v2f __attribute__((ext_vector_type(2)));
typedef float v8f __attribute__((ext_vector_type(8)));

// ---------------------------------------------------------------------------
// Pass 1: per-row windowed correlation via sliding windows over LDS tiles.
// grid = ROWS*8 blocks, 256 threads; each thread produces 16 consecutive cols.
// ---------------------------------------------------------------------------
__global__ __launch_bounds__(256) void corr_kernel(const float* __restrict__ x,
                                                   float* __restrict__ corrBuf)
{
    __shared__ float s1[SKMAX];
    __shared__ float s2[SKMAX];

    const int tid   = threadIdx.x;
    const int row   = blockIdx.x >> 3;
    const int tile0 = (blockIdx.x & 7) * TILE;

    const float* g1 = x + (size_t)(row * 2) * N + tile0;
    const float* g2 = g1 + N;

    const int rem   = N - tile0;
    const int nLoad = (NLOAD < rem) ? NLOAD : rem;   // 4195, or 4096 on last tile

#ifdef USE_ASYNC_LDS
    // CDNA5 async global->LDS: data flows memory -> LDS without VGPR staging.
    for (int i = tid; i < nLoad; i += 256) {
        __builtin_amdgcn_global_load_async_to_lds_b32(
            (g_int*)(g1 + i), (l_int*)&s1[SK(i)], 0, 0);
        __builtin_amdgcn_global_load_async_to_lds_b32(
            (g_int*)(g2 + i), (l_int*)&s2[SK(i)], 0, 0);
    }
#if __has_builtin(__builtin_amdgcn_s_wait_asynccnt)
    __builtin_amdgcn_s_wait_asynccnt(0);
#else
    asm volatile("s_wait_asynccnt 0" ::: "memory");
#endif
#else
    for (int i = tid; i < nLoad; i += 256) {
        s1[SK(i)] = g1[i];
        s2[SK(i)] = g2[i];
    }
#endif
    __syncthreads();

    const int   o    = tid * 16;        // local output base for this thread
    const float invw = 1.0f / (float)WIN;

    // Build the initial 100-wide window sums.
    float sa = 0.f, sb = 0.f, saa = 0.f, sbb = 0.f, sab = 0.f;
#pragma unroll 4
    for (int k = 0; k < WIN; ++k) {
        const float a = s1[SK(o + k)];
        const float b = s2[SK(o + k)];
        sa += a; sb += b;
        saa = fmaf(a, a, saa);
        sbb = fmaf(b, b, sbb);
        sab = fmaf(a, b, sab);
    }

    float* crow = corrBuf + (size_t)row * STRIDE;
#pragma unroll
    for (int j = 0; j < 16; ++j) {
        const int c = tile0 + o + j;
        if (c < OUTC) {
            const float cov = sab - sa * sb * invw;
            const float v1  = saa - sa * sa * invw;
            const float v2  = sbb - sb * sb * invw;
            crow[c] = cov / (sqrtf(v1) * sqrtf(v2));
        }
        if (j < 15) {   // slide window by one
            const float aN = s1[SK(o + j + WIN)];
            const float bN = s2[SK(o + j + WIN)];
            const float aO = s1[SK(o + j)];
            const float bO = s2[SK(o + j)];
            sa  += aN - aO;
            sb  += bN - bO;
            saa += aN * aN - aO * aO;
            sbb += bN * bN - bO * bO;
            sab += aN * bN - aO * bO;
        }
    }
}

// ---------------------------------------------------------------------------
// Pass 2: column mean over 512 rows as a WMMA reduction.
// A = 16x4 all-ones, B = 4 rows x 16 cols of corr, D accumulates column sums.
// Each wave32 handles 16 columns with 128 accumulating v_wmma_f32_16x16x4_f32.
// ---------------------------------------------------------------------------
__global__ __launch_bounds__(256) void avg_wmma_kernel(const float* __restrict__ corrBuf,
                                                       float* __restrict__ avg)
{
#if defined(__gfx1250__)
    const int lane = threadIdx.x & 31;
    const int wave = threadIdx.x >> 5;
    const int g    = blockIdx.x * 8 + wave;          // 16-column group id
    const int ngrp = (OUTC + 15) / 16;               // 2042
    if (g >= ngrp) return;                            // wave-uniform: EXEC stays all-ones

    const int col  = g * 16 + (lane & 15);
    const int rsel = (lane >> 4) << 1;               // lanes 0-15 -> rows k,k+1 ; 16-31 -> k+2,k+3

    v2f ones; ones.x = 1.0f; ones.y = 1.0f;          // A(16x4) = 1 in every slot
    v8f acc = {};

    const float* p = corrBuf + (size_t)rsel * STRIDE + col;
#pragma unroll 4
    for (int r = 0; r < ROWS; r += 4) {
        v2f bmat;
        bmat.x = p[0];                               // B row K = rsel
        bmat.y = p[STRIDE];                          // B row K = rsel+1
        acc = __builtin_amdgcn_wmma_f32_16x16x4_f32(
            /*neg_a=*/false, ones, /*neg_b=*/false, bmat,
            /*c_mod=*/(short)0, acc, /*reuse_a=*/false, /*reuse_b=*/false);
        p += (size_t)4 * STRIDE;
    }

    // D row M=0 lives in acc[0] of lanes 0-15 (N = lane): the full column sum.
    if (lane < 16 && col < OUTC) avg[col] = acc[0] * (1.0f / (float)ROWS);
#endif
}

// ---------------------------------------------------------------------------
// Pass 3: out = relu(corr - avg), coalesced; corr + avg come from L2.
// ---------------------------------------------------------------------------
__global__ __launch_bounds__(256) void finalize_kernel(const float* __restrict__ corrBuf,
                                                       const float* __restrict__ avg,
                                                       float* __restrict__ out)
{
    const int c   = blockIdx.x * 256 + threadIdx.x;
    const int row = blockIdx.y;
    if (c < OUTC) {
        const float v = corrBuf[(size_t)row * STRIDE + c] - avg[c];
        out[(size_t)row * OUTC + c] = (v > 0.0f) ? v : 0.0f;
    }
}

extern "C" void kernel_launch(void* const* d_in, const int* in_sizes, int n_in,
                              void* d_out, int out_size, void* d_ws, size_t ws_size,
                              hipStream_t stream)
{
    (void)in_sizes; (void)n_in; (void)out_size; (void)ws_size;

    const float* x   = (const float*)d_in[0];
    float*       out = (float*)d_out;

    // Workspace layout: corr[512][32672] then avg[32672+] ; ~64 MiB total.
    float* corrBuf = (float*)d_ws;
    float* avg     = corrBuf + (size_t)ROWS * STRIDE;

    corr_kernel<<<dim3(ROWS * 8), 256, 0, stream>>>(x, corrBuf);
    avg_wmma_kernel<<<dim3(256), 256, 0, stream>>>(corrBuf, avg);
    finalize_kernel<<<dim3((OUTC + 255) / 256, ROWS), 256, 0, stream>>>(corrBuf, avg, out);
}